// MODEL_25984552141438
// MI455X (gfx1250) — compile-verified
//
#include <hip/hip_runtime.h>

// ---------------------------------------------------------------------------
// LightGCN (2-layer) propagation for MI455X / gfx1250.
// Memory-bound COO scatter-add pipeline:
//   - hardware fp32 scatter atomics via inline `global_atomic_add_f32
//     scope:SCOPE_DEV` (RMW resolves in the 192MB L2; the 128MB accumulator
//     is L2-resident)
//   - final residual blend expressed as exact fp32 identity-matmuls via
//     V_WMMA_F32_16X16X4_F32.
// ---------------------------------------------------------------------------

#define USER_NUM 200000
#define ITEM_NUM 300000
#define NNODES   (USER_NUM + ITEM_NUM)   // 500000
#define DIM      64
#define NEDGES   1000000

static_assert(NNODES % 16 == 0, "tile rows");
static_assert(USER_NUM % 16 == 0, "user/item boundary must be tile aligned");

typedef __attribute__((ext_vector_type(2))) float v2f;
typedef __attribute__((ext_vector_type(8))) float v8f;

// Hardware float atomic-add, no-return form (tracked by STOREcnt; s_endpgm's
// implicit wait-idle drains it). Device scope -> RMW performed at L2.
__device__ __forceinline__ void gpu_atomic_add_f32(float* p, float v) {
    asm volatile("global_atomic_add_f32 %0, %1, off scope:SCOPE_DEV"
                 :
                 : "v"(p), "v"(v)
                 : "memory");
}

// ---------------- 1) zero scratch (degrees + aggregation buffer) -----------
__global__ __launch_bounds__(256) void k_zero(float4* __restrict__ p, int n4) {
    int i = blockIdx.x * blockDim.x + threadIdx.x;
    if (i < n4) {
        float4 z; z.x = 0.f; z.y = 0.f; z.z = 0.f; z.w = 0.f;
        p[i] = z;
    }
}

// ---------------- 2) degree accumulation (float atomics, counts exact) -----
__global__ __launch_bounds__(256) void k_degrees(const int* __restrict__ src,
                                                 const int* __restrict__ dst,
                                                 float* __restrict__ deg_out,
                                                 float* __restrict__ deg_in) {
    int e = blockIdx.x * blockDim.x + threadIdx.x;
    if (e < NEDGES) {
        gpu_atomic_add_f32(&deg_out[src[e]], 1.0f);
        gpu_atomic_add_f32(&deg_in[dst[e]], 1.0f);
    }
}

// ---------------- 3) deg -> deg^-0.5 in place ------------------------------
__global__ __launch_bounds__(256) void k_norms(float* __restrict__ deg, int n) {
    int i = blockIdx.x * blockDim.x + threadIdx.x;
    if (i < n) deg[i] = rsqrtf(fmaxf(deg[i], 1.0f));
}

// ---------------- 4/6) scatter: agg[dst] += x[src] * out_norm[src] ---------
// 64 threads (2 waves) per edge -> one contiguous 256B row gather, 64 L2-
// resident hardware float atomics into the 128MB agg buffer.
__global__ __launch_bounds__(256) void k_scatter(const int* __restrict__ src,
                                                 const int* __restrict__ dst,
                                                 const float* __restrict__ x0,
                                                 const float* __restrict__ x1,
                                                 int split,
                                                 const float* __restrict__ norm_out,
                                                 float* __restrict__ agg) {
    int t = blockIdx.x * blockDim.x + threadIdx.x;
    int e = t >> 6;
    int j = t & 63;
    if (e < NEDGES) {
        int s = src[e];
        int d = dst[e];
        float scale = norm_out[s];
        const float* row = (s < split) ? (x0 + (size_t)s * DIM)
                                       : (x1 + (size_t)(s - split) * DIM);
        float v = row[j] * scale;
        gpu_atomic_add_f32(&agg[(size_t)d * DIM + j], v);
    }
}

// ---------------- 5) e1 = agg * in_norm -> stage in d_out; re-zero agg -----
__global__ __launch_bounds__(256) void k_finalize1(float* __restrict__ agg,
                                                   const float* __restrict__ norm_in,
                                                   float* __restrict__ e1) {
    int i = blockIdx.x * blockDim.x + threadIdx.x;          // [0, N*16) float4s
    if (i < NNODES * (DIM / 4)) {
        float nin = norm_in[i >> 4];
        float4 v = ((float4*)agg)[i];
        v.x *= nin; v.y *= nin; v.z *= nin; v.w *= nin;
        ((float4*)e1)[i] = v;
        float4 z; z.x = 0.f; z.y = 0.f; z.z = 0.f; z.w = 0.f;
        ((float4*)agg)[i] = z;                               // ready for layer 2
    }
}

// ---------------- 7) out = base + 0.5*e1 + (1/3)*(agg*in_norm) -------------
// One wave per 16x16 tile. Two chained V_WMMA_F32_16X16X4_F32 identity
// matmuls per K-chunk (scale folded into the 4x16 identity slices); exact in
// fp32 (every product is either s*x or 0).
// A layout (16x4 f32): lane L(0-15): K=0,1 ; lane L+16: K=2,3 (row = L&15).
// B layout (4x16 f32): VGPR0: K=0 (lanes 0-15) / K=2 (lanes 16-31); VGPR1: K=1/K=3.
// C/D layout (16x16 f32): VGPR v: M=v (lanes 0-15) / M=v+8 (lanes 16-31).
__global__ __launch_bounds__(256) void k_finalize2_wmma(
        const float* __restrict__ user_emb,
        const float* __restrict__ item_emb,
        const float* __restrict__ e1,        // staged in d_out
        const float* __restrict__ agg,       // layer-2 aggregation
        const float* __restrict__ norm_in,
        float* __restrict__ out) {
    const int lane = threadIdx.x & 31;
    const int wave = threadIdx.x >> 5;
    const int tile = blockIdx.x * (blockDim.x >> 5) + wave;  // exact cover, EXEC all-1
    const int node0 = (tile >> 2) << 4;                      // 16-node group
    const int c0    = (tile & 3) << 4;                       // 16-dim column block
    const int nn = lane & 15;
    const int hi = lane >> 4;

    // Tile never straddles the user/item boundary (USER_NUM % 16 == 0).
    const float* base = (node0 < USER_NUM)
                            ? (user_emb + (size_t)node0 * DIM)
                            : (item_emb + (size_t)(node0 - USER_NUM) * DIM);

    // C accumulator = base tile.
    v8f c;
#pragma unroll
    for (int v = 0; v < 8; ++v) {
        int row = v + 8 * hi;
        c[v] = base[(size_t)row * DIM + c0 + nn];
    }

    const float nin = norm_in[node0 + nn];                   // A-row for this lane
    const size_t arow = (size_t)(node0 + nn) * DIM + c0;

#pragma unroll
    for (int g = 0; g < 4; ++g) {
        const int off = g * 4 + 2 * hi;                      // K columns for this lane
        // D += E1 x (0.5 * I)
        v2f a1; a1[0] = e1[arow + off]; a1[1] = e1[arow + off + 1];
        v2f b1; b1[0] = (nn == off)     ? 0.5f : 0.0f;
                b1[1] = (nn == off + 1) ? 0.5f : 0.0f;
        c = __builtin_amdgcn_wmma_f32_16x16x4_f32(false, a1, false, b1,
                                                  (short)0, c, false, false);
        // D += (agg * in_norm) x ((1/3) * I)
        v2f a2; a2[0] = agg[arow + off] * nin; a2[1] = agg[arow + off + 1] * nin;
        v2f b2; b2[0] = (nn == off)     ? (1.0f / 3.0f) : 0.0f;
                b2[1] = (nn == off + 1) ? (1.0f / 3.0f) : 0.0f;
        c = __builtin_amdgcn_wmma_f32_16x16x4_f32(false, a2, false, b2,
                                                  (short)0, c, false, false);
    }

    // Store D. In-place over e1 is safe: each tile reads only itself, and all
    // reads (data-dependence through c) precede the stores.
    float* o = out + (size_t)node0 * DIM;
#pragma unroll
    for (int v = 0; v < 8; ++v) {
        int row = v + 8 * hi;
        o[(size_t)row * DIM + c0 + nn] = c[v];
    }
}

// ---------------------------------------------------------------------------
extern "C" void kernel_launch(void* const* d_in, const int* in_sizes, int n_in,
                              void* d_out, int out_size, void* d_ws, size_t ws_size,
                              hipStream_t stream) {
    const float* user_emb = (const float*)d_in[0];
    const float* item_emb = (const float*)d_in[1];
    const int*   src      = (const int*)d_in[2];
    const int*   dst      = (const int*)d_in[3];
    float* out = (float*)d_out;

    // Workspace layout (floats): norm_out[N] | norm_in[N] | agg[N*64]  (132 MB)
    float* wsf      = (float*)d_ws;
    float* norm_out = wsf;                 // doubles as deg_out accumulator
    float* norm_in  = wsf + NNODES;        // doubles as deg_in accumulator
    float* agg      = wsf + 2 * NNODES;

    // 1) zero degrees + agg: (2N + 64N) floats = 66N -> float4 count
    {
        int n4 = (66 * NNODES) / 4;                       // 8,250,000
        int blocks = (n4 + 255) / 256;
        k_zero<<<blocks, 256, 0, stream>>>((float4*)wsf, n4);
    }
    // 2) degrees
    k_degrees<<<(NEDGES + 255) / 256, 256, 0, stream>>>(src, dst, norm_out, norm_in);
    // 3) norms (both arrays, contiguous 2N)
    k_norms<<<(2 * NNODES + 255) / 256, 256, 0, stream>>>(norm_out, 2 * NNODES);
    // 4) layer-1 scatter: gather from input embeddings
    {
        long threads = (long)NEDGES * 64;                 // 64M
        int blocks = (int)((threads + 255) / 256);        // 250,000
        k_scatter<<<blocks, 256, 0, stream>>>(src, dst, user_emb, item_emb,
                                              USER_NUM, norm_out, agg);
    }
    // 5) e1 = agg*in_norm -> d_out (staging); agg re-zeroed for layer 2
    k_finalize1<<<(NNODES * (DIM / 4) + 255) / 256, 256, 0, stream>>>(agg, norm_in, out);
    // 6) layer-2 scatter: gather from staged e1 (all rows in one buffer)
    {
        long threads = (long)NEDGES * 64;
        int blocks = (int)((threads + 255) / 256);
        k_scatter<<<blocks, 256, 0, stream>>>(src, dst, out, out,
                                              NNODES, norm_out, agg);
    }
    // 7) WMMA blend: out = base + 0.5*e1 + (1/3)*(agg*in_norm)
    {
        int tiles = (NNODES / 16) * (DIM / 16);           // 125,000
        int blocks = tiles / 8;                           // 8 waves/block, exact
        k_finalize2_wmma<<<blocks, 256, 0, stream>>>(user_emb, item_emb,
                                                     out, agg, norm_in, out);
    }
}